// VitNormOutput_73023033967040
// MI455X (gfx1250) — compile-verified
//
#include <hip/hip_runtime.h>
#include <hip/hip_bf16.h>

// ---------------------------------------------------------------------------
// ViT norm-output attribution, fully fused for MI455X (gfx1250, wave32).
// B=2 H=12 S=197 D=768 Hd=64.  Store-bandwidth bound (~243MB out @ 23.3TB/s).
// Matrix math: V_WMMA_F32_16X16X4_F32.  probs gather: TENSOR_LOAD_TO_LDS (TDM).
// ---------------------------------------------------------------------------

typedef __attribute__((ext_vector_type(2))) float v2f;
typedef __attribute__((ext_vector_type(8))) float v8f;
typedef __attribute__((ext_vector_type(4))) unsigned int v4u;
typedef __attribute__((ext_vector_type(8))) unsigned int v8u;

#define B_  2
#define H_  12
#define S_  197
#define D_  768
#define HD_ 64
#define KPAD 208        // 13*16, padded k-range for A-matrix tiles
#define LN_EPS 1e-12f

__device__ __forceinline__ float waveReduceSum(float v) {
#pragma unroll
  for (int o = 16; o > 0; o >>= 1) v += __shfl_xor(v, o, 32);
  return v;
}

// 256-thread (8-wave) block reduce; returns total to all threads.
__device__ __forceinline__ float blockReduceSum(float v, float* red) {
  int lane = threadIdx.x & 31, w = threadIdx.x >> 5;
  v = waveReduceSum(v);
  __syncthreads();
  if (lane == 0) red[w] = v;
  __syncthreads();
  float r = red[0];
#pragma unroll
  for (int i = 1; i < 8; ++i) r += red[i];
  return r;
}

// f32 WMMA 16x16x4 fragment conventions (ISA 7.12.2):
//  A(16x4):  vgpr0 = A[l][2*half], vgpr1 = A[l][2*half+1]   (l=lane&15, half=lane>>4)
//  B(4x16):  vgpr0 = B[2*half][l], vgpr1 = B[2*half+1][l]
//  C(16x16): c[r]  = C[r+8*half][l]
// Out-of-range M rows / N cols: addresses are clamped (no value masking) —
// garbage only lands in C rows/cols that are never stored.

// ---------------------------------------------------------------------------
// K1: transformed[b,h,s,d] = sum_v value[b,h,s,v] * dense_weight[d, h*64+v]
// One wave per 16x16 tile: M=s(13 tiles) N=d(48 tiles) K=64.  24*624 waves.
// All 16 A/B fragment pairs loaded up-front, then a single WMMA chain.
// ---------------------------------------------------------------------------
__global__ __launch_bounds__(256) void k_transformed(
    const float* __restrict__ value, const float* __restrict__ W,
    float* __restrict__ T) {
  int wave = threadIdx.x >> 5, lane = threadIdx.x & 31;
  int half = lane >> 4, l = lane & 15;
  long gw = (long)blockIdx.x * 8 + wave;
  if (gw >= (long)B_ * H_ * 13 * 48) return;      // wave-uniform
  int nt = gw % 48; gw /= 48;
  int mt = gw % 13; gw /= 13;
  int h  = gw % H_; int b = gw / H_;
  int s0 = mt * 16, d0 = nt * 16;

  const float* Vb = value + ((long)b * H_ + h) * S_ * HD_;   // [s][v]
  const float* Wc = W + (long)(d0 + l) * D_ + h * HD_;       // column d0+l
  int mrow = s0 + l;
  bool mval = (mrow < S_);
  const float* Arow = Vb + (long)(mval ? mrow : 0) * HD_;    // clamped addr

  v2f af[16], bf[16];
#pragma unroll
  for (int t = 0; t < 16; ++t) {
    int ka = 4 * t + 2 * half;
    af[t].x = Arow[ka]; af[t].y = Arow[ka + 1];
  }
#pragma unroll
  for (int t = 0; t < 16; ++t) {
    int ka = 4 * t + 2 * half;
    bf[t].x = Wc[ka]; bf[t].y = Wc[ka + 1];
  }
  v8f c = {};
#pragma unroll
  for (int t = 0; t < 16; ++t)
    c = __builtin_amdgcn_wmma_f32_16x16x4_f32(false, af[t], false, bf[t],
                                              (short)0, c, false, false);
  float* Tb = T + ((long)b * H_ + h) * S_ * D_;
#pragma unroll
  for (int r = 0; r < 8; ++r) {
    int row = s0 + r + 8 * half;
    if (row < S_) Tb[(long)row * D_ + d0 + l] = c[r];
  }
}

// ---------------------------------------------------------------------------
// K1b: tnorm[b,h,s] = ||transformed[b,h,s,:]||   (one wave per row, b128 loads)
// ---------------------------------------------------------------------------
__global__ __launch_bounds__(256) void k_tnorm(const float* __restrict__ T,
                                               float* __restrict__ tnorm) {
  int wave = threadIdx.x >> 5, lane = threadIdx.x & 31;
  long row = (long)blockIdx.x * 8 + wave;
  if (row >= (long)B_ * H_ * S_) return;
  const float4* p4 = (const float4*)(T + row * D_);
  float acc = 0.f;
  for (int q = lane; q < D_ / 4; q += 32) {
    float4 v = p4[q];
    acc += v.x * v.x + v.y * v.y + v.z * v.z + v.w * v.w;
  }
  acc = waveReduceSum(acc);
  if (lane == 0) tnorm[row] = sqrtf(acc);
}

// ---------------------------------------------------------------------------
// K2: weighted_norm[b,h,k,s] = |probs[b,h,k,s]| * tnorm[b,h,s]
// ---------------------------------------------------------------------------
__global__ __launch_bounds__(256) void k_wnorm(const float* __restrict__ probs,
                                               const float* __restrict__ tnorm,
                                               float* __restrict__ out) {
  long i = (long)blockIdx.x * 256 + threadIdx.x;
  if (i >= (long)B_ * H_ * S_ * S_) return;
  long s = i % S_;
  long bh = i / ((long)S_ * S_);
  out[i] = fabsf(probs[i]) * tnorm[bh * S_ + s];
}

// ---------------------------------------------------------------------------
// K3: psum[b,k,d] = sum_{h,s} probs[b,h,k,s]*transformed[b,h,s,d]
// GEMM per b: M=d(48 tiles) N=k(13 tiles) K=H*S=2364.  2*48*13=1248 waves.
// A[m=d][kk=hs] = T (linear in kk);  B[kk][n=k] = probs (clamped column).
// ---------------------------------------------------------------------------
__global__ __launch_bounds__(256) void k_psum(const float* __restrict__ probs,
                                              const float* __restrict__ T,
                                              float* __restrict__ psum) {
  int wave = threadIdx.x >> 5, lane = threadIdx.x & 31;
  int half = lane >> 4, l = lane & 15;
  int gw = blockIdx.x * 8 + wave;
  if (gw >= B_ * 48 * 13) return;                 // wave-uniform
  int ntk = gw % 13; gw /= 13;
  int mtd = gw % 48; int b = gw / 48;
  int d0 = mtd * 16, k0 = ntk * 16;

  const float* Tb = T + (long)b * H_ * S_ * D_;         // [hs][d]
  const float* Pb = probs + (long)b * H_ * S_ * S_;     // [h][k][s]
  int ncol = k0 + l;
  bool nval = ncol < S_;
  long pk = (long)(nval ? ncol : 0) * S_;               // clamped addr
  __builtin_prefetch(Tb + d0, 0, 1);

  v8f c = {};
  const int HS = H_ * S_;                               // 2364 = 197*12
  for (int kk = 0; kk < HS; kk += 12) {                 // 3 WMMAs per batch
    v2f a3[3], b3[3];
#pragma unroll
    for (int t = 0; t < 3; ++t) {
      int kb = kk + 4 * t + 2 * half;
      long abase = (long)kb * D_ + d0 + l;
      a3[t].x = Tb[abase]; a3[t].y = Tb[abase + D_];
      int h0 = kb / S_,       sx0 = kb - h0 * S_;
      int h1 = (kb + 1) / S_, sx1 = (kb + 1) - h1 * S_;
      b3[t].x = Pb[(long)h0 * S_ * S_ + pk + sx0];
      b3[t].y = Pb[(long)h1 * S_ * S_ + pk + sx1];
    }
#pragma unroll
    for (int t = 0; t < 3; ++t)
      c = __builtin_amdgcn_wmma_f32_16x16x4_f32(false, a3[t], false, b3[t],
                                                (short)0, c, false, false);
  }
  float* pw = psum + (long)b * S_ * D_;
#pragma unroll
  for (int r = 0; r < 8; ++r) {
    int drow = d0 + r + 8 * half;
    if (nval) pw[(long)ncol * D_ + drow] = c[r];
  }
}

// ---------------------------------------------------------------------------
// K4: per (b,k): rstd, diagonal vectors, closed-form mixing ratios.
//  sum_s summed = psum;  sum_s rw = psum+hidden;
//  sum_s post   = lnw*(psum+hid - mean_d(psum+hid))*rstd  (LN affine in d)
// ---------------------------------------------------------------------------
__global__ __launch_bounds__(256) void k_mix(
    const float* __restrict__ probs, const float* __restrict__ T,
    const float* __restrict__ psum, const float* __restrict__ hidden,
    const float* __restrict__ lnw, const float* __restrict__ pre_ln,
    float* __restrict__ rstd_ws, float* __restrict__ out_attn,
    float* __restrict__ out_res, float* __restrict__ out_ln) {
  __shared__ float red[8];
  __shared__ float pdiag[H_];
  int b = blockIdx.x / S_, k = blockIdx.x - b * S_;
  int tid = threadIdx.x;

  const float* pl = pre_ln + ((long)b * S_ + k) * D_;
  float s1 = 0.f, s2 = 0.f;
  for (int d = tid; d < D_; d += 256) { float v = pl[d]; s1 += v; s2 += v * v; }
  s1 = blockReduceSum(s1, red);
  s2 = blockReduceSum(s2, red);
  float m = s1 * (1.f / D_);
  float rstd = rsqrtf(s2 * (1.f / D_) - m * m + LN_EPS);
  if (tid == 0) rstd_ws[blockIdx.x] = rstd;
  if (tid < H_) pdiag[tid] = probs[(((long)b * H_ + tid) * S_ + k) * S_ + k];
  __syncthreads();

  float sd[3], ps[3], rw[3], prw[3];
  float srw = 0.f, sprw = 0.f;
#pragma unroll
  for (int j = 0; j < 3; ++j) {
    int d = tid + 256 * j;
    float acc = 0.f;
#pragma unroll
    for (int h = 0; h < H_; ++h)
      acc += pdiag[h] * T[(((long)b * H_ + h) * S_ + k) * D_ + d];
    sd[j] = acc;                                  // summed diagonal
    ps[j] = psum[((long)b * S_ + k) * D_ + d];
    float hv = hidden[((long)b * S_ + k) * D_ + d];
    rw[j] = acc + hv;                             // rw diagonal
    prw[j] = ps[j] + hv;                          // sum_s rw
    srw += rw[j]; sprw += prw[j];
  }
  srw  = blockReduceSum(srw, red);
  sprw = blockReduceSum(sprw, red);
  float mean_rw = srw * (1.f / D_), mean_prw = sprw * (1.f / D_);

  float n_sd = 0.f, n_mix = 0.f, n_rw = 0.f, n_pd = 0.f, n_pmix = 0.f;
#pragma unroll
  for (int j = 0; j < 3; ++j) {
    int d = tid + 256 * j;
    float lw = lnw[d];
    float postd = (rw[j]  - mean_rw)  * lw * rstd;
    float posts = (prw[j] - mean_prw) * lw * rstd;
    float mixv  = ps[j] - sd[j];
    float pmix  = posts - postd;
    n_sd += sd[j] * sd[j];   n_mix += mixv * mixv;  n_rw += rw[j] * rw[j];
    n_pd += postd * postd;   n_pmix += pmix * pmix;
  }
  n_sd  = blockReduceSum(n_sd, red);
  n_mix = blockReduceSum(n_mix, red);
  n_rw  = blockReduceSum(n_rw, red);
  n_pd  = blockReduceSum(n_pd, red);
  n_pmix = blockReduceSum(n_pmix, red);
  if (tid == 0) {
    float mn = sqrtf(n_mix);
    out_attn[blockIdx.x] = mn / (mn + sqrtf(n_sd));
    out_res [blockIdx.x] = mn / (mn + sqrtf(n_rw));
    float mnl = sqrtf(n_pmix);
    out_ln  [blockIdx.x] = mnl / (mnl + sqrtf(n_pd));
  }
}

// ---------------------------------------------------------------------------
// K5: fused main pass, one block per (b,s).
//  - probs gather (12 x 208 tile, strides 197 / 38809) via ONE Tensor Data
//    Mover descriptor; TDM OOB reads return zero => free k>=197 padding.
//  - summed tile via WMMA (K=12), B-frags held in registers across k-tiles.
//  - LN + norms with float4 LDS reads, post_ln rows stored as b128.
// ---------------------------------------------------------------------------
__global__ __launch_bounds__(256) void k_fused(
    const float* __restrict__ probs, const float* __restrict__ T,
    const float* __restrict__ hidden, const float* __restrict__ lnw,
    const float* __restrict__ rstd_ws, float* __restrict__ out_snorm,
    float* __restrict__ out_rwnorm, float* __restrict__ out_pnorm,
    float* __restrict__ out_post) {
  extern __shared__ float smem[];
  float* p_lds = smem;                    // [12][KPAD]   A-matrix (TDM dest)
  float* tile  = p_lds + H_ * KPAD;       // [16][768]    summed tile
  float* lnw_l = tile + 16 * D_;          // [768]
  float* hid_l = lnw_l + D_;              // [768]
  float* red   = hid_l + D_;              // [8+]

  int b = blockIdx.x / S_, s = blockIdx.x - b * S_;
  int tid = threadIdx.x, wave = tid >> 5, lane = tid & 31;
  int half = lane >> 4, l = lane & 15;

  // ---- TDM: p_lds[h][k] = probs[b,h,k,s]; x=1 elem, y=208 (zero-pad >=197),
  //      z=12.  data_size=4B, stride_y=197, stride_z=38809 elements. ----
  unsigned ldsoff = (unsigned)(unsigned long long)(__SIZE_TYPE__)p_lds;
  unsigned long long gaddr =
      (unsigned long long)(__SIZE_TYPE__)(probs + (long)b * H_ * S_ * S_ + s);
  v4u g0 = { 1u,                                   // count=1, user mode
             ldsoff,                               // lds_addr
             (unsigned)gaddr,                      // global_addr[31:0]
             ((unsigned)(gaddr >> 32) & 0x01FFFFFFu) | (2u << 30) }; // +type=2
  v8u g1 = { 0x00020000u,                          // data_size=4B
             (1u << 16),                           // tensor_dim0 = 1
             (197u << 16),                         // tensor_dim1 = 197
             (1u << 16),                           // tile_dim0 = 1
             208u | (12u << 16),                   // tile_dim1=208 tile_dim2=12
             197u,                                 // tensor_dim0_stride = 197
             (38809u << 16),                       // tensor_dim1_stride = 38809
             0u };
  v4u g2 = { 12u, 1u, 0u, 0u };                    // tensor_dim2=12, dim3=1
  v4u g3 = { 0u, 0u, 0u, 0u };
  if (tid < 32) {                                  // one wave issues the DMA
    asm volatile("tensor_load_to_lds %0, %1, %2, %3"
                 :: "s"(g0), "s"(g1), "s"(g2), "s"(g3)
                 : "memory");
    __builtin_amdgcn_s_wait_tensorcnt(0);
  }

  { // lnw + hidden row -> LDS (b128)
    const float4* lnwg = (const float4*)lnw;
    const float4* hidg = (const float4*)(hidden + ((long)b * S_ + s) * D_);
    float4* lnwl4 = (float4*)lnw_l;
    float4* hidl4 = (float4*)hid_l;
    for (int q = tid; q < D_ / 4; q += 256) { lnwl4[q] = lnwg[q]; hidl4[q] = hidg[q]; }
  }

  // B-fragments: wave owns d-tiles [wave*6, wave*6+6); reused for all k-tiles.
  v2f bfr[6][3];
#pragma unroll
  for (int j = 0; j < 6; ++j) {
    int d0 = (wave * 6 + j) * 16;
#pragma unroll
    for (int t = 0; t < 3; ++t) {
      int kk = 4 * t + 2 * half;                      // h index
      bfr[j][t].x = T[(((long)b * H_ + kk)     * S_ + s) * D_ + d0 + l];
      bfr[j][t].y = T[(((long)b * H_ + kk + 1) * S_ + s) * D_ + d0 + l];
    }
  }
  float c0p = 0.f;
  for (int d = tid; d < D_; d += 256) { float lw = lnw[d]; c0p += lw * lw; }
  float C0 = blockReduceSum(c0p, red);                // sum lnw^2
  __syncthreads();                                    // TDM + LDS fills visible

  for (int kt = 0; kt < 13; ++kt) {
    int k0 = kt * 16;
    // --- WMMA: summed[k0..k0+15, wave*96..+96) into LDS tile ---
#pragma unroll
    for (int j = 0; j < 6; ++j) {
      int d0 = (wave * 6 + j) * 16;
      v8f c = {};
#pragma unroll
      for (int t = 0; t < 3; ++t) {
        int kk = 4 * t + 2 * half;
        v2f a;
        a.x = p_lds[kk * KPAD + k0 + l];
        a.y = p_lds[(kk + 1) * KPAD + k0 + l];
        c = __builtin_amdgcn_wmma_f32_16x16x4_f32(false, a, false, bfr[j][t],
                                                  (short)0, c, false, false);
      }
#pragma unroll
      for (int r = 0; r < 8; ++r)
        tile[(r + 8 * half) * D_ + d0 + l] = c[r];
    }
    __syncthreads();
    // --- row processing: each wave handles 2 rows, float4 everywhere ---
#pragma unroll
    for (int rr = 0; rr < 2; ++rr) {
      int rloc = wave * 2 + rr;
      int k = k0 + rloc;
      if (k < S_) {                                   // wave-uniform
        float dm = (k == s) ? 1.f : 0.f;              // diagonal residual mask
        float rstd = rstd_ws[b * S_ + k];
        const float4* tv = (const float4*)(tile + rloc * D_);
        const float4* hv4 = (const float4*)hid_l;
        const float4* lv4 = (const float4*)lnw_l;
        float a_s2 = 0.f, a_rw = 0.f, a_rw2 = 0.f, a_l1 = 0.f, a_l2 = 0.f;
        for (int q = lane; q < D_ / 4; q += 32) {
          float4 sv = tv[q], hh = hv4[q], lw = lv4[q];
          float rwx = sv.x + dm * hh.x, rwy = sv.y + dm * hh.y;
          float rwz = sv.z + dm * hh.z, rww = sv.w + dm * hh.w;
          float lx2 = lw.x * lw.x, ly2 = lw.y * lw.y;
          float lz2 = lw.z * lw.z, lw2 = lw.w * lw.w;
          a_s2 += sv.x * sv.x + sv.y * sv.y + sv.z * sv.z + sv.w * sv.w;
          a_rw += rwx + rwy + rwz + rww;
          a_rw2 += rwx * rwx + rwy * rwy + rwz * rwz + rww * rww;
          a_l1 += lx2 * rwx + ly2 * rwy + lz2 * rwz + lw2 * rww;
          a_l2 += lx2 * rwx * rwx + ly2 * rwy * rwy + lz2 * rwz * rwz +
                  lw2 * rww * rww;
        }
        a_s2 = waveReduceSum(a_s2);
        a_rw = waveReduceSum(a_rw);
        a_rw2 = waveReduceSum(a_rw2);
        a_l1 = waveReduceSum(a_l1);
        a_l2 = waveReduceSum(a_l2);
        float mean = a_rw * (1.f / D_);
        float pn2 = a_l2 - 2.f * mean * a_l1 + mean * mean * C0;
        long nidx = ((long)b * S_ + k) * S_ + s;
        if (lane == 0) {
          out_snorm[nidx]  = sqrtf(a_s2);
          out_rwnorm[nidx] = sqrtf(a_rw2);
          out_pnorm[nidx]  = rstd * sqrtf(fmaxf(pn2, 0.f));
        }
        float4* orow = (float4*)(out_post + nidx * (long)D_);
        for (int q = lane; q < D_ / 4; q += 32) {     // 512B-coalesced stores
          float4 sv = tv[q], hh = hv4[q], lw = lv4[q];
          float4 o;
          o.x = (sv.x + dm * hh.x - mean) * lw.x * rstd;
          o.y = (sv.y + dm * hh.y - mean) * lw.y * rstd;
          o.z = (sv.z + dm * hh.z - mean) * lw.z * rstd;
          o.w = (sv.w + dm * hh.w - mean) * lw.w * rstd;
          orow[q] = o;
        }
      }
    }
    __syncthreads();
  }
}

// ---------------------------------------------------------------------------
extern "C" void kernel_launch(void* const* d_in, const int* in_sizes, int n_in,
                              void* d_out, int out_size, void* d_ws,
                              size_t ws_size, hipStream_t stream) {
  const float* hidden = (const float*)d_in[0];   // (B,S,D)
  const float* probs  = (const float*)d_in[1];   // (B,H,S,S)
  const float* value  = (const float*)d_in[2];   // (B,H,S,Hd)
  const float* W      = (const float*)d_in[3];   // (D,D)
  const float* lnw    = (const float*)d_in[4];   // (D,)
  const float* pre_ln = (const float*)d_in[5];   // (B,S,D)
  float* out = (float*)d_out;

  // workspace layout (floats)
  float* T     = (float*)d_ws;                       // B*H*S*D = 3,631,104
  float* tnorm = T + (long)B_ * H_ * S_ * D_;        // 4,728
  float* psum  = tnorm + (long)B_ * H_ * S_;         // B*S*D = 302,592
  float* rstd  = psum + (long)B_ * S_ * D_;          // 394

  // output offsets (return order)
  const long N_WN = (long)B_ * H_ * S_ * S_;         // 931,416
  const long N_SS = (long)B_ * S_ * S_;              // 77,618
  const long N_PL = N_SS * D_;                       // 59,610,624
  float* o_wn    = out;
  float* o_snorm = o_wn + N_WN;
  float* o_rwn   = o_snorm + N_SS;
  float* o_pn    = o_rwn + N_SS;
  float* o_post  = o_pn + N_SS;
  float* o_attn  = o_post + N_PL;
  float* o_res   = o_attn + (long)B_ * S_;
  float* o_ln    = o_res + (long)B_ * S_;

  k_transformed<<<(B_ * H_ * 13 * 48) / 8, 256, 0, stream>>>(value, W, T);
  k_tnorm<<<(B_ * H_ * S_ + 7) / 8, 256, 0, stream>>>(T, tnorm);
  k_wnorm<<<(int)((N_WN + 255) / 256), 256, 0, stream>>>(probs, tnorm, o_wn);
  k_psum<<<(B_ * 48 * 13) / 8, 256, 0, stream>>>(probs, T, psum);
  k_mix<<<B_ * S_, 256, 0, stream>>>(probs, T, psum, hidden, lnw, pre_ln,
                                     rstd, o_attn, o_res, o_ln);
  const int smem_bytes = (H_ * KPAD + 16 * D_ + D_ + D_ + 16) * sizeof(float);
  k_fused<<<B_ * S_, 256, smem_bytes, stream>>>(probs, T, hidden, lnw, rstd,
                                                o_snorm, o_rwn, o_pn, o_post);
}